// VectorQuantize2_78572131713244
// MI455X (gfx1250) — compile-verified
//
#include <hip/hip_runtime.h>
#include <hip/hip_bf16.h>

typedef __attribute__((ext_vector_type(2))) float v2f;
typedef __attribute__((ext_vector_type(8))) float v8f;
typedef int v4i __attribute__((vector_size(16)));   // matches builtin param type

#define AS1 __attribute__((address_space(1)))
#define AS3 __attribute__((address_space(3)))

#if __has_builtin(__builtin_amdgcn_global_load_async_to_lds_b128)
#define HAVE_ASYNC_LDS 1
#define ASYNC_B128(gsrc, ldst) \
    __builtin_amdgcn_global_load_async_to_lds_b128((AS1 v4i*)(gsrc), (AS3 v4i*)(ldst), 0, 0)
#else
#define HAVE_ASYNC_LDS 0
#endif

#if __has_builtin(__builtin_amdgcn_s_wait_asynccnt)
#define WAIT_ASYNC0() __builtin_amdgcn_s_wait_asynccnt(0)
#else
#define WAIT_ASYNC0() asm volatile("s_wait_asynccnt 0x0" ::: "memory")
#endif

#define C_DIM      256
#define K_CODES    8192
#define N_ROWS     16384
#define ROWS_PER_WG 64
#define CHUNK      32
#define NCHUNK     (K_CODES / CHUNK)
#define LDSX       260          // padded row stride (floats): stride%64 == 4 -> conflict-free column reads
#define XQ_ELEMS   4194304      // 16*256*32*32
#define LOSS_IDX   4194304
#define CODE_IDX   4194305
#define BETA_F     0.25f

// --------------------------------------------------------------------------
// Kernel A: enorm2[k] = sum_c weight[k][c]^2   (one block per code row)
// --------------------------------------------------------------------------
__global__ __launch_bounds__(256) void vq_enorm(const float* __restrict__ w,
                                                float* __restrict__ en) {
    const int row = blockIdx.x;
    const int t   = threadIdx.x;
    float v = w[row * C_DIM + t];
    v *= v;
    #pragma unroll
    for (int off = 16; off; off >>= 1) v += __shfl_xor(v, off, 32);
    __shared__ float s[8];
    if ((t & 31) == 0) s[t >> 5] = v;
    __syncthreads();
    if (t == 0) {
        float tot = 0.0f;
        #pragma unroll
        for (int i = 0; i < 8; ++i) tot += s[i];
        en[row] = tot;
    }
}

// --------------------------------------------------------------------------
// Kernel B: fused distance-GEMM (fp32 WMMA) + argmin + gather + transpose +
//           per-block loss partial.
// Code tiles ping-pong in LDS via GLOBAL_LOAD_ASYNC_TO_LDS (ASYNCcnt),
// 2 independent WMMA accumulator chains.
// Grid: 256 blocks (64 xf rows each), 256 threads = 8 waves (4 M x 2 N).
// --------------------------------------------------------------------------
__global__ __launch_bounds__(256) void vq_main(const float* __restrict__ x,
                                               const float* __restrict__ weight,
                                               const float* __restrict__ enorm2g,
                                               float* __restrict__ out,
                                               float* __restrict__ partial) {
    __shared__ float sX[ROWS_PER_WG * LDSX];    // 64 x-rows, padded
    __shared__ float sE[2][CHUNK * LDSX];       // ping-pong 32-code tiles
    __shared__ float sRedV[2 * ROWS_PER_WG];
    __shared__ int   sRedI[2 * ROWS_PER_WG];
    __shared__ int   sCode[ROWS_PER_WG];
    __shared__ float sWsum[8];

    const int t        = threadIdx.x;
    const int lane     = t & 31;
    const int w        = t >> 5;
    const int wave_m   = w & 3;        // which 16-row M tile
    const int wave_n   = w >> 2;       // which 16-code half of the chunk
    const int row_base = blockIdx.x * ROWS_PER_WG;
    const int bI       = row_base >> 10;      // batch index (64 | 1024)
    const int pix_base = row_base & 1023;     // h*32+w base within image

    // ---- stage 64 xf rows into LDS; xf[row][c] = x[b*262144 + c*1024 + pix]
    {
        const int row = t & 63;
        const int c0  = t >> 6;
        const float* xsrc = x + bI * (C_DIM * 1024) + pix_base + row;
        #pragma unroll 8
        for (int c = c0; c < C_DIM; c += 4)
            sX[row * LDSX + c] = xsrc[c << 10];
    }

    const int er = t >> 3;   // code row (0..31) this thread stages
    const int ej = t & 7;    // 16B segment within row

    // ---- prologue: stage chunk 0 into buffer 0
#if HAVE_ASYNC_LDS
    {
        const float* wsrc = weight + er * C_DIM + ej * 4;
        float* edst = &sE[0][er * LDSX + ej * 4];
        #pragma unroll
        for (int cc = 0; cc < 8; ++cc)
            ASYNC_B128(wsrc + cc * 32, edst + cc * 32);
        WAIT_ASYNC0();
    }
#else
    {
        const float* wsrc = weight + er * C_DIM + ej * 4;
        float* edst = &sE[0][er * LDSX + ej * 4];
        #pragma unroll
        for (int cc = 0; cc < 8; ++cc)
            *(float4*)(edst + cc * 32) = *(const float4*)(wsrc + cc * 32);
    }
#endif
    __syncthreads();

    // ---- per-lane running argmin state (8 accumulator rows per lane)
    float bestv[8];
    int   besti[8];
    #pragma unroll
    for (int r = 0; r < 8; ++r) { bestv[r] = 3.402823466e+38f; besti[r] = 0; }

    // A fragment base (ISA 32-bit 16x4 layout: lane = M + 16*(K/2), reg = K&1)
    const float* aPtr = sX + (wave_m * 16 + (lane & 15)) * LDSX + 2 * (lane >> 4);

    for (int j = 0; j < NCHUNK; ++j) {
        const int  buf  = j & 1;
        const bool more = (j + 1) < NCHUNK;

#if HAVE_ASYNC_LDS
        // kick off async DMA of the next code tile straight into LDS;
        // latency hidden behind the 64 WMMAs below (no staging VGPRs).
        if (more) {
            const float* wsrc = weight + ((j + 1) * CHUNK + er) * C_DIM + ej * 4;
            float* edst = &sE[buf ^ 1][er * LDSX + ej * 4];
            #pragma unroll
            for (int cc = 0; cc < 8; ++cc)
                ASYNC_B128(wsrc + cc * 32, edst + cc * 32);
            if (j + 2 < NCHUNK)
                __builtin_prefetch(weight + ((j + 2) * CHUNK + er) * C_DIM + ej * 32, 0, 1);
        }
#else
        float4 stg[8];
        if (more) {
            const float* wsrc = weight + ((j + 1) * CHUNK + er) * C_DIM + ej * 4;
            #pragma unroll
            for (int cc = 0; cc < 8; ++cc)
                stg[cc] = *(const float4*)(wsrc + cc * 32);
            if (j + 2 < NCHUNK)
                __builtin_prefetch(weight + ((j + 2) * CHUNK + er) * C_DIM + ej * 32, 0, 1);
        }
#endif

        // S(16x16) = X(16x256) * E_chunk^T : 64 WMMAs on 2 independent chains
        const float* bPtr = &sE[buf][(wave_n * 16 + (lane & 15)) * LDSX + 2 * (lane >> 4)];
        v8f acc0 = {};
        v8f acc1 = {};
        #pragma unroll
        for (int kk = 0; kk < 64; kk += 2) {
            v2f a0 = *(const v2f*)(aPtr + 4 * kk);
            v2f b0 = *(const v2f*)(bPtr + 4 * kk);
            v2f a1 = *(const v2f*)(aPtr + 4 * kk + 4);
            v2f b1 = *(const v2f*)(bPtr + 4 * kk + 4);
            acc0 = __builtin_amdgcn_wmma_f32_16x16x4_f32(
                false, a0, false, b0, (short)0, acc0, false, false);
            acc1 = __builtin_amdgcn_wmma_f32_16x16x4_f32(
                false, a1, false, b1, (short)0, acc1, false, false);
        }

        // score_k = ||e_k||^2 - 2*(x . e_k); D-frag: n = lane&15 (all regs)
        const int   cg = j * CHUNK + wave_n * 16 + (lane & 15);
        const float en = enorm2g[cg];
        #pragma unroll
        for (int r = 0; r < 8; ++r) {
            const float s = fmaf(-2.0f, acc0[r] + acc1[r], en);
            if (s < bestv[r]) { bestv[r] = s; besti[r] = cg; }
        }

#if HAVE_ASYNC_LDS
        if (more) WAIT_ASYNC0();     // own wave's DMA done; barrier publishes tile
#else
        if (more) {
            float* edst = &sE[buf ^ 1][er * LDSX + ej * 4];
            #pragma unroll
            for (int cc = 0; cc < 8; ++cc)
                *(float4*)(edst + cc * 32) = stg[cc];
        }
#endif
        __syncthreads();
    }

    // ---- reduce across the 16 lanes of each D-fragment half (same M row)
    #pragma unroll
    for (int r = 0; r < 8; ++r) {
        float v = bestv[r];
        int   i = besti[r];
        #pragma unroll
        for (int off = 8; off; off >>= 1) {
            const float ov = __shfl_xor(v, off, 16);
            const int   oi = __shfl_xor(i, off, 16);
            if (ov < v || (ov == v && oi < i)) { v = ov; i = oi; }
        }
        if ((lane & 15) == 0) {
            const int m   = r + ((lane >> 4) << 3);   // lanes 0-15: M=r, 16-31: M=r+8
            const int row = wave_m * 16 + m;
            sRedV[wave_n * ROWS_PER_WG + row] = v;
            sRedI[wave_n * ROWS_PER_WG + row] = i;
        }
    }
    __syncthreads();

    // ---- combine the two code halves, emit code_out (as float)
    if (t < ROWS_PER_WG) {
        const float v0 = sRedV[t];
        const float v1 = sRedV[ROWS_PER_WG + t];
        const int   i0 = sRedI[t];
        const int   i1 = sRedI[ROWS_PER_WG + t];
        const int code = (v1 < v0 || (v1 == v0 && i1 < i0)) ? i1 : i0;
        sCode[t] = code;
        out[CODE_IDX + row_base + t] = (float)code;
    }
    __syncthreads();

    // ---- gather x_q = emb[code], write NCHW (contiguous 256B per (c,quad)),
    //      accumulate sum((x_q - xf)^2)
    float lsum = 0.0f;
    {
        const int row  = t & 63;
        const int c0   = t >> 6;
        const int code = sCode[row];
        const float* wrow = weight + code * C_DIM;
        float* outp = out + bI * (C_DIM * 1024) + pix_base + row;
        #pragma unroll 8
        for (int c = c0; c < C_DIM; c += 4) {
            const float q  = wrow[c];
            const float xv = sX[row * LDSX + c];
            const float d  = q - xv;
            lsum = fmaf(d, d, lsum);
            outp[c << 10] = q;
        }
    }
    #pragma unroll
    for (int off = 16; off; off >>= 1) lsum += __shfl_xor(lsum, off, 32);
    if (lane == 0) sWsum[w] = lsum;
    __syncthreads();
    if (t == 0) {
        float tot = 0.0f;
        #pragma unroll
        for (int i = 0; i < 8; ++i) tot += sWsum[i];
        partial[blockIdx.x] = tot;   // deterministic: no atomics
    }
}

// --------------------------------------------------------------------------
// Kernel C: deterministic fixed-order loss reduction
// --------------------------------------------------------------------------
__global__ __launch_bounds__(256) void vq_finalize(const float* __restrict__ partial,
                                                   float* __restrict__ out) {
    __shared__ float s[256];
    const int t = threadIdx.x;
    s[t] = partial[t];
    __syncthreads();
    for (int off = 128; off; off >>= 1) {
        if (t < off) s[t] += s[t + off];
        __syncthreads();
    }
    if (t == 0)
        out[LOSS_IDX] = (1.0f + BETA_F) * s[0] / (float)XQ_ELEMS;
}

// --------------------------------------------------------------------------
extern "C" void kernel_launch(void* const* d_in, const int* in_sizes, int n_in,
                              void* d_out, int out_size, void* d_ws, size_t ws_size,
                              hipStream_t stream) {
    (void)in_sizes; (void)n_in; (void)out_size; (void)ws_size;
    const float* x      = (const float*)d_in[0];   // (16,256,32,32) f32
    const float* weight = (const float*)d_in[1];   // (8193,256) f32
    float* out     = (float*)d_out;                // xq(4194304) | loss(1) | code(16384)
    float* enorm   = (float*)d_ws;                 // 8192 floats
    float* partial = enorm + K_CODES;              // 256 floats

    vq_enorm   <<<K_CODES, 256, 0, stream>>>(weight, enorm);
    vq_main    <<<N_ROWS / ROWS_PER_WG, 256, 0, stream>>>(x, weight, enorm, out, partial);
    vq_finalize<<<1, 256, 0, stream>>>(partial, out);
}